// Pool2d_8727373545577
// MI455X (gfx1250) — compile-verified
//
#include <hip/hip_runtime.h>
#include <math.h>

// 3x3 max pool, stride 2, pad 1 (fill 0.0), dilation 1.
// X: (32,192,224,224) f32 -> Y: (32,192,112,112) f32.  Memory-bound (~1.54 GB
// traffic -> ~66us floor at 23.3 TB/s).
// Strategy: one TDM tensor_load_to_lds DMA per workgroup stages a 33x224 f32
// tile into LDS (fallback: per-lane async global->LDS b128 copies); compute
// uses separable rolling horizontal-max from LDS; non-temporal coalesced
// stores for the write-once output stream.

#define H_IN     224
#define W_IN     224
#define H_OUT    112
#define W_OUT    112
#define OH_TILE  16
#define IN_ROWS  (2 * OH_TILE + 1)   // 33 input rows per tile
#define ROW_F4   (W_IN / 4)          // 56 float4 chunks per row
#define NTHREADS 224                 // 7 waves; one output column per thread

typedef unsigned int       u32;
typedef unsigned long long u64;
typedef __attribute__((ext_vector_type(4))) u32 u32x4;
typedef __attribute__((ext_vector_type(4))) int i32x4;
typedef __attribute__((ext_vector_type(8))) int i32x8;

#if defined(__has_builtin)
#if __has_builtin(__builtin_amdgcn_tensor_load_to_lds)
#define HAVE_TDM 1
#endif
#endif
#ifndef HAVE_TDM
#define HAVE_TDM 0
#endif

__global__ __launch_bounds__(NTHREADS)
void maxpool3x3s2_kernel(const float* __restrict__ X, float* __restrict__ Y)
{
    __shared__ float smem[IN_ROWS * W_IN];   // 33*224*4 = 29568 B -> ~9 WGs/WGP

    const int plane    = blockIdx.y;              // n*C + c  (0..6143)
    const int oh0      = blockIdx.x * OH_TILE;    // first output row of tile
    const int ih0      = oh0 * 2 - 1;             // -1 only when blockIdx.x==0
    const int tid      = threadIdx.x;
    const int pad_top  = (ih0 < 0) ? 1 : 0;       // one padded row at the very top
    const int nrows    = IN_ROWS - pad_top;       // rows actually fetched
    const int ih_start = ih0 + pad_top;           // first valid input row

    const float*  plane_base = X + (size_t)plane * (H_IN * W_IN);
    const unsigned lds_base  = (unsigned)(size_t)(void*)smem; // low32 of flat LDS ptr = LDS byte offset

    if (pad_top) {
        // LDS row 0 represents the ih=-1 pad row; reference pads with 0.0f.
        for (int c4 = tid; c4 < ROW_F4; c4 += NTHREADS)
            *(float4*)&smem[c4 * 4] = make_float4(0.f, 0.f, 0.f, 0.f);
    }

#if HAVE_TDM
    // ---- One Tensor-Data-Mover DMA stages the whole nrows x 224 tile ----
    if (tid == 0) {
        const u64 gaddr   = (u64)(size_t)(plane_base + (size_t)ih_start * W_IN);
        const u32 lds_dst = lds_base + (u32)(pad_top * W_IN * 4);

        // D# group 0: [1:0]=count=1, [63:32]=lds_addr, [120:64]=global_addr, [127:126]=type(2)
        u32x4 g0;
        g0.x = 1u;
        g0.y = lds_dst;
        g0.z = (u32)gaddr;
        g0.w = ((u32)(gaddr >> 32) & 0x01FFFFFFu) | 0x80000000u;

        // D# group 1: data_size=4B, tensor_dim0=224, tensor_dim1=nrows,
        //             tile_dim0=224, tile_dim1=nrows, tensor_dim0_stride=224
        i32x8 g1;
        g1[0] = (int)(2u << 16);              // [17:16] data_size = 2 -> 4 bytes
        g1[1] = (int)(224u << 16);            // [79:48] tensor_dim0 = 224 (low half)
        g1[2] = (int)((u32)nrows << 16);      // [111:80] tensor_dim1 = nrows (low half)
        g1[3] = (int)(224u << 16);            // [127:112] tile_dim0 = 224
        g1[4] = nrows;                        // [143:128] tile_dim1 = nrows
        g1[5] = 224;                          // [207:160] tensor_dim0_stride = 224
        g1[6] = 0;
        g1[7] = 0;

        const i32x4 gz = {0, 0, 0, 0};        // groups 2/3: dims/tiles unused (2D)
#if __clang_major__ >= 23
        const i32x8 gz8 = {0, 0, 0, 0, 0, 0, 0, 0};
        __builtin_amdgcn_tensor_load_to_lds(g0, g1, gz, gz, gz8, 0);
#else
        __builtin_amdgcn_tensor_load_to_lds(g0, g1, gz, gz, 0);
#endif
        __builtin_amdgcn_s_wait_tensorcnt(0); // wave 0 drains its TDM op
    }
#else
    // ---- Fallback: per-lane async global->LDS b128 copies ----
    for (int k = tid; k < IN_ROWS * ROW_F4; k += NTHREADS) {
        const int r  = k / ROW_F4;            // LDS row 0..32
        const int c4 = k - r * ROW_F4;
        const int ih = ih0 + r;
        const unsigned lds_addr = lds_base + (unsigned)(k * 16);
        if (ih >= 0) {                        // bottom never overflows (max ih = 223)
            const unsigned long long ga =
                (unsigned long long)(size_t)(plane_base + (size_t)ih * W_IN + c4 * 4);
            asm volatile("global_load_async_to_lds_b128 %0, %1, off"
                         : : "v"(lds_addr), "v"(ga) : "memory");
        }
    }
    asm volatile("s_wait_asynccnt 0x0" ::: "memory");
#endif
    __syncthreads();

    // ---- Compute: separable max with rolling horizontal-max reuse ----
    // Thread -> output column ow, rows [oh0+8g, oh0+8g+8).
    const int ow    = tid % W_OUT;
    const int g     = tid / W_OUT;            // 0 or 1
    const int c     = 2 * ow - 1;             // -1..221 (c+2 <= 223 always)
    const int rbase = 16 * g;                 // first LDS row for this thread

    float* out_col = Y + (size_t)plane * (H_OUT * W_OUT)
                       + (size_t)(oh0 + 8 * g) * W_OUT + ow;

    auto hmax = [&](int r) -> float {
        const float* row = &smem[r * W_IN];
        const float a = (c >= 0) ? row[c] : 0.0f;   // left pad contributes 0.0
        return fmaxf(a, fmaxf(row[c + 1], row[c + 2]));
    };

    float hA = hmax(rbase);        // h(2*oh_local)
    float hB = hmax(rbase + 1);    // h(2*oh_local+1)
    #pragma unroll
    for (int j = 0; j < 8; ++j) {
        const float hC = hmax(rbase + 2 * j + 2);
        const float m  = fmaxf(hA, fmaxf(hB, hC));
        // Output is written once and never re-read: stream it past the caches.
        __builtin_nontemporal_store(m, &out_col[(size_t)j * W_OUT]);
        hA = hC;                                   // reused by next output row
        if (j < 7) hB = hmax(rbase + 2 * j + 3);
    }
}

extern "C" void kernel_launch(void* const* d_in, const int* in_sizes, int n_in,
                              void* d_out, int out_size, void* d_ws, size_t ws_size,
                              hipStream_t stream)
{
    (void)in_sizes; (void)n_in; (void)out_size; (void)d_ws; (void)ws_size;
    const float* X = (const float*)d_in[0];
    float*       Y = (float*)d_out;

    dim3 grid(H_OUT / OH_TILE, 32 * 192);   // (7, 6144)
    maxpool3x3s2_kernel<<<grid, NTHREADS, 0, stream>>>(X, Y);
}